// TemporalGNN_3229815407314
// MI455X (gfx1250) — compile-verified
//
#include <hip/hip_runtime.h>
#include <math.h>

#define F_DIM 8
#define T_DIM 12
#define FT    96
#define OC    32

typedef __attribute__((ext_vector_type(2))) float v2f;
typedef __attribute__((ext_vector_type(8))) float v8f;
typedef __attribute__((ext_vector_type(4))) unsigned int v4u;
typedef __attribute__((ext_vector_type(8))) int v8i;
typedef __attribute__((ext_vector_type(4))) int v4i;

__device__ __forceinline__ v8f splat8(float v) {
    v8f r;
#pragma unroll
    for (int i = 0; i < 8; i++) r[i] = v;
    return r;
}

// D = A(16x4 f32) * B(4x16 f32) + C(16x16 f32), full fp32 matrix pipe
__device__ __forceinline__ v8f wmma4(v2f a, v2f b, v8f c) {
    return __builtin_amdgcn_wmma_f32_16x16x4_f32(false, a, false, b, (short)0, c, false, false);
}

__device__ __forceinline__ float sigmoidf(float x) { return 1.0f / (1.0f + __expf(-x)); }

// ---------------- small prep kernels ----------------

__global__ void k_deg_init(float* deg, int n) {
    int i = blockIdx.x * blockDim.x + threadIdx.x;
    if (i < n) deg[i] = 1.0f;   // self-loop weight
}

__global__ void k_deg_accum(const int* __restrict__ dst, const float* __restrict__ w,
                            float* __restrict__ deg, int e) {
    int i = blockIdx.x * blockDim.x + threadIdx.x;
    if (i < e) atomicAdd(&deg[dst[i]], w[i]);
}

__global__ void k_dis(float* deg, int n) {
    int i = blockIdx.x * blockDim.x + threadIdx.x;
    if (i < n) deg[i] = rsqrtf(deg[i]);   // deg >= 1 always (self loop)
}

// fw = sigmoid(x @ mlp_w + b);  wx[n,f,t] = x[n,f,t] * fw[n,t]
__global__ void k_fwx(const float* __restrict__ x, const float* __restrict__ mlp_w,
                      const float* __restrict__ mlp_b, float* __restrict__ wx, int n) {
    __shared__ float sw[FT * T_DIM];
    __shared__ float sb[T_DIM];
    for (int k = threadIdx.x; k < FT * T_DIM; k += blockDim.x) sw[k] = mlp_w[k];
    if (threadIdx.x < T_DIM) sb[threadIdx.x] = mlp_b[threadIdx.x];
    __syncthreads();
    int node = blockIdx.x * blockDim.x + threadIdx.x;
    if (node >= n) return;
    const float* xp = x + (size_t)node * FT;
    float acc[T_DIM];
#pragma unroll
    for (int t = 0; t < T_DIM; t++) acc[t] = sb[t];
    for (int k = 0; k < FT; k++) {
        float xv = xp[k];
#pragma unroll
        for (int t = 0; t < T_DIM; t++) acc[t] += xv * sw[k * T_DIM + t];
    }
#pragma unroll
    for (int t = 0; t < T_DIM; t++) acc[t] = sigmoidf(acc[t]);
    float* wp = wx + (size_t)node * FT;
    for (int f = 0; f < F_DIM; f++)
#pragma unroll
        for (int t = 0; t < T_DIM; t++) wp[f * T_DIM + t] = xp[f * T_DIM + t] * acc[t];
}

// Agg init with self-loop contribution: norm_self = dis[i]^2
__global__ void k_selfinit(const float* __restrict__ wx, const float* __restrict__ dis,
                           float* __restrict__ agg, int n) {
    long long gid = (long long)blockIdx.x * blockDim.x + threadIdx.x;
    if (gid >= (long long)n * FT) return;
    int i = (int)(gid / FT);
    float d = dis[i];
    agg[gid] = wx[gid] * d * d;
}

// one wave per edge: Agg[dst] += wx[src] * (dis[src]*w*dis[dst]); 96 comps, 3 per lane
__global__ void k_edge(const int* __restrict__ src, const int* __restrict__ dst,
                       const float* __restrict__ w, const float* __restrict__ dis,
                       const float* __restrict__ wx, float* __restrict__ agg, int e) {
    int wv   = (blockIdx.x * blockDim.x + threadIdx.x) >> 5;
    int lane = threadIdx.x & 31;
    if (wv >= e) return;
    int s = src[wv], d = dst[wv];
    float nm = dis[s] * w[wv] * dis[d];
    const float* xs = wx + (size_t)s * FT;
    float* ad = agg + (size_t)d * FT;
#pragma unroll
    for (int r = 0; r < 3; r++) {
        int k = lane + 32 * r;
        atomicAdd(&ad[k], xs[k] * nm);
    }
}

// ---------------- WMMA GRU kernel ----------------

#define GRU_THREADS 128
#define WPB 4            // waves per block
#define NPW 16           // nodes per wave tile
#define NPB 64           // nodes per block

// B operand, weight W row-major [K x ncol]: tile picks cols 16t..16t+15, chunk c rows 4c..4c+3
__device__ __forceinline__ v2f bop(const float* W, int ncol, int tile, int c, int lane_n, int khalf) {
    int nn = 16 * tile + lane_n;
    int k  = 4 * c + khalf;
    v2f b; b[0] = W[k * ncol + nn]; b[1] = W[(k + 1) * ncol + nn]; return b;
}

// A operand from staged 16x32 row-major tile
__device__ __forceinline__ v2f aop_stage(const float* S, int c, int mrow, int khalf) {
    int k = 4 * c + khalf;
    v2f a; a[0] = S[mrow * OC + k]; a[1] = S[mrow * OC + k + 1]; return a;
}

// write a 16x32 matrix (two C/D-layout tiles) to staging (row-major m*32+n)
__device__ __forceinline__ void stage_write(float* S, const v8f* M, int lane) {
    int mbase  = (lane >> 4) * 8;
    int lane_n = lane & 15;
#pragma unroll
    for (int t = 0; t < 2; t++)
#pragma unroll
        for (int r = 0; r < 8; r++)
            S[(mbase + r) * OC + 16 * t + lane_n] = M[t][r];
}

// conv: C(16x32) = AggLDS_slice(16x8) @ cw(8x32) + cb ; A operand read from LDS agg tile
__device__ __forceinline__ void conv16(v8f C[2], const float* aggrow, int p,
                                       const float* cw, const float* cb,
                                       int khalf, int lane_n) {
    v2f a[2];
#pragma unroll
    for (int cc = 0; cc < 2; cc++) {
        int f = 4 * cc + khalf;
        a[cc][0] = aggrow[f * T_DIM + p];
        a[cc][1] = aggrow[(f + 1) * T_DIM + p];
    }
#pragma unroll
    for (int t = 0; t < 2; t++) {
        v8f c = splat8(cb[16 * t + lane_n]);
#pragma unroll
        for (int cc = 0; cc < 2; cc++)
            c = wmma4(a[cc], bop(cw, OC, t, cc, lane_n, khalf), c);
        C[t] = c;
    }
}

// C(16x32) += Staged(16x32) @ W(32x32)
__device__ __forceinline__ void gemm_acc(v8f C[2], const float* S, const float* W,
                                         int lane_n, int mrow, int khalf) {
#pragma unroll
    for (int t = 0; t < 2; t++) {
        v8f c = C[t];
#pragma unroll
        for (int cc = 0; cc < 8; cc++)
            c = wmma4(aop_stage(S, cc, mrow, khalf), bop(W, OC, t, cc, lane_n, khalf), c);
        C[t] = c;
    }
}

__global__ void __launch_bounds__(GRU_THREADS) k_gru(
    const float* __restrict__ agg, const float* __restrict__ att,
    const float* __restrict__ czw, const float* __restrict__ czb,
    const float* __restrict__ lzw, const float* __restrict__ lzb,
    const float* __restrict__ crw, const float* __restrict__ crb,
    const float* __restrict__ lrw, const float* __restrict__ lrb,
    const float* __restrict__ chw, const float* __restrict__ chb,
    const float* __restrict__ lhw, const float* __restrict__ lhb,
    const float* __restrict__ outw, const float* __restrict__ outb,
    float* __restrict__ out, int n) {
    __shared__ float s_agg[NPB * FT];     // 64 nodes x 96 feats = 24.6 KB, TDM target
    __shared__ float s_cw[3][F_DIM * OC];
    __shared__ float s_lw[3][2 * OC * OC];
    __shared__ float s_cb[3][OC];
    __shared__ float s_lb[3][OC];
    __shared__ float s_ow[OC * 16];       // out_w padded 32x16
    __shared__ float s_ob[16];
    __shared__ float s_prob[T_DIM];
    __shared__ float s_stage[WPB][NPW * OC];

    const int tid  = threadIdx.x;
    const int wave = tid >> 5;
    const int lane = tid & 31;
    const int base = blockIdx.x * NPB;

    // ---- Tensor Data Mover: DMA the 64x96 f32 Agg tile into LDS (zero-fill past n) ----
#if defined(__gfx1250__) && __has_builtin(__builtin_amdgcn_tensor_load_to_lds)
    if (wave == 0) {
        unsigned long long gaddr = (unsigned long long)(uintptr_t)(agg + (size_t)base * FT);
        unsigned lds_off = (unsigned)(uintptr_t)(void*)s_agg;   // LDS aperture: low 32 bits = LDS offset
        unsigned rows_avail = (unsigned)(n - base);             // OOB rows read as zero

        v4u g0;
        g0[0] = 1u;                                             // count=1, is_restore=0, no gather
        g0[1] = lds_off;                                        // lds_addr
        g0[2] = (unsigned)(gaddr & 0xFFFFFFFFull);              // global_addr[31:0]
        g0[3] = (unsigned)((gaddr >> 32) & 0x01FFFFFFull) | (2u << 30);  // addr[56:32] | type=2

        v8i g1;
        g1[0] = (int)(2u << 16);                                // data_size=4B; no barrier/iterate/pad
        g1[1] = (int)((unsigned)FT << 16);                      // tensor_dim0[15:0] in [31:16]
        g1[2] = (int)(rows_avail << 16);                        // dim0 hi=0 | tensor_dim1 lo in [31:16]
        g1[3] = (int)((rows_avail >> 16) | ((unsigned)FT << 16));// tensor_dim1 hi | tile_dim0=96
        g1[4] = (int)NPB;                                       // tile_dim1=64, tile_dim2=0
        g1[5] = (int)FT;                                        // tensor_dim0_stride[31:0]=96
        g1[6] = 0;                                              // stride hi, dim1_stride lo
        g1[7] = 0;

        v4i g2 = {0, 0, 0, 0};
        v4i g3 = {0, 0, 0, 0};
        v8i g4 = {0, 0, 0, 0, 0, 0, 0, 0};                      // extra group (6-arg builtin form)
        __builtin_amdgcn_tensor_load_to_lds(g0, g1, g2, g3, g4, 0);
        __builtin_amdgcn_s_wait_tensorcnt(0);
    }
#else
    for (int k = tid; k < NPB * FT; k += GRU_THREADS) {
        int node = base + k / FT;
        if (node >= n) node = n - 1;
        s_agg[k] = agg[(size_t)node * FT + (k % FT)];
    }
#endif

    // ---- stage weights / biases / softmax(att) ----
    for (int k = tid; k < F_DIM * OC; k += GRU_THREADS) {
        s_cw[0][k] = czw[k]; s_cw[1][k] = crw[k]; s_cw[2][k] = chw[k];
    }
    for (int k = tid; k < 2 * OC * OC; k += GRU_THREADS) {
        s_lw[0][k] = lzw[k]; s_lw[1][k] = lrw[k]; s_lw[2][k] = lhw[k];
    }
    if (tid < OC) {
        s_cb[0][tid] = czb[tid]; s_cb[1][tid] = crb[tid]; s_cb[2][tid] = chb[tid];
        s_lb[0][tid] = lzb[tid]; s_lb[1][tid] = lrb[tid]; s_lb[2][tid] = lhb[tid];
    }
    for (int k = tid; k < OC * 16; k += GRU_THREADS) {
        int oc = k >> 4, t = k & 15;
        s_ow[k] = (t < T_DIM) ? outw[oc * T_DIM + t] : 0.0f;
    }
    if (tid < 16) s_ob[tid] = (tid < T_DIM) ? outb[tid] : 0.0f;
    if (tid == 0) {  // softmax(att), T=12
        float m = att[0];
        for (int t = 1; t < T_DIM; t++) m = fmaxf(m, att[t]);
        float ssum = 0.0f, ee[T_DIM];
        for (int t = 0; t < T_DIM; t++) { ee[t] = __expf(att[t] - m); ssum += ee[t]; }
        for (int t = 0; t < T_DIM; t++) s_prob[t] = ee[t] / ssum;
    }
    __syncthreads();

    const int lane_n = lane & 15;
    const int mrow   = lane & 15;
    const int khalf  = (lane >> 4) << 1;   // 0 or 2
    float* stage = s_stage[wave];
    const float* aggrow = s_agg + (size_t)(wave * NPW + mrow) * FT;   // this lane's node row in LDS

    v8f H[2], Hacc[2];
#pragma unroll
    for (int t = 0; t < 2; t++) { H[t] = splat8(0.0f); Hacc[t] = splat8(0.0f); }

#pragma unroll 1
    for (int p = 0; p < T_DIM; p++) {
        v8f tmp[2], gz[2], gr[2], gh[2], Z[2], R[2];

        // z conv -> stage -> gz top half
        conv16(tmp, aggrow, p, s_cw[0], s_cb[0], khalf, lane_n);
        __syncthreads(); stage_write(stage, tmp, lane); __syncthreads();
        gz[0] = splat8(s_lb[0][lane_n]); gz[1] = splat8(s_lb[0][16 + lane_n]);
        gemm_acc(gz, stage, s_lw[0], lane_n, mrow, khalf);

        // r conv -> stage -> gr top half
        conv16(tmp, aggrow, p, s_cw[1], s_cb[1], khalf, lane_n);
        __syncthreads(); stage_write(stage, tmp, lane); __syncthreads();
        gr[0] = splat8(s_lb[1][lane_n]); gr[1] = splat8(s_lb[1][16 + lane_n]);
        gemm_acc(gr, stage, s_lw[1], lane_n, mrow, khalf);

        // H -> stage -> gz/gr bottom halves
        __syncthreads(); stage_write(stage, H, lane); __syncthreads();
        gemm_acc(gz, stage, s_lw[0] + OC * OC, lane_n, mrow, khalf);
        gemm_acc(gr, stage, s_lw[1] + OC * OC, lane_n, mrow, khalf);
#pragma unroll
        for (int t = 0; t < 2; t++)
#pragma unroll
            for (int r = 0; r < 8; r++) { Z[t][r] = sigmoidf(gz[t][r]); R[t][r] = sigmoidf(gr[t][r]); }

        // h conv -> stage -> gh top half
        conv16(tmp, aggrow, p, s_cw[2], s_cb[2], khalf, lane_n);
        __syncthreads(); stage_write(stage, tmp, lane); __syncthreads();
        gh[0] = splat8(s_lb[2][lane_n]); gh[1] = splat8(s_lb[2][16 + lane_n]);
        gemm_acc(gh, stage, s_lw[2], lane_n, mrow, khalf);

        // H*R -> stage -> gh bottom half
#pragma unroll
        for (int t = 0; t < 2; t++)
#pragma unroll
            for (int r = 0; r < 8; r++) tmp[t][r] = H[t][r] * R[t][r];
        __syncthreads(); stage_write(stage, tmp, lane); __syncthreads();
        gemm_acc(gh, stage, s_lw[2] + OC * OC, lane_n, mrow, khalf);

        // GRU update + attention accumulation
        float pb = s_prob[p];
#pragma unroll
        for (int t = 0; t < 2; t++)
#pragma unroll
            for (int r = 0; r < 8; r++) {
                float ht = tanhf(gh[t][r]);
                float z  = Z[t][r];
                H[t][r]  = z * H[t][r] + (1.0f - z) * ht;
                Hacc[t][r] += pb * H[t][r];
            }
    }

    // out = relu(Hacc) @ out_w + out_b  (B padded 32x16, cols >= 12 are zero)
    v8f Ar[2];
#pragma unroll
    for (int t = 0; t < 2; t++)
#pragma unroll
        for (int r = 0; r < 8; r++) Ar[t][r] = fmaxf(Hacc[t][r], 0.0f);
    __syncthreads(); stage_write(stage, Ar, lane); __syncthreads();
    v8f o = splat8(s_ob[lane_n]);
#pragma unroll
    for (int cc = 0; cc < 8; cc++)
        o = wmma4(aop_stage(stage, cc, mrow, khalf), bop(s_ow, 16, 0, cc, lane_n, khalf), o);

    int mbase = (lane >> 4) * 8;
#pragma unroll
    for (int r = 0; r < 8; r++) {
        int node = base + wave * NPW + mbase + r;
        if (node < n && lane_n < T_DIM) out[(size_t)node * T_DIM + lane_n] = o[r];
    }
}

// ---------------- launcher ----------------

extern "C" void kernel_launch(void* const* d_in, const int* in_sizes, int n_in,
                              void* d_out, int out_size, void* d_ws, size_t ws_size,
                              hipStream_t stream) {
    const float* x    = (const float*)d_in[0];
    const int*   ei   = (const int*)d_in[1];
    const float* ew   = (const float*)d_in[2];
    const float* mlpw = (const float*)d_in[3];
    const float* mlpb = (const float*)d_in[4];
    const float* att  = (const float*)d_in[5];
    const float* czw  = (const float*)d_in[6];
    const float* czb  = (const float*)d_in[7];
    const float* lzw  = (const float*)d_in[8];
    const float* lzb  = (const float*)d_in[9];
    const float* crw  = (const float*)d_in[10];
    const float* crb  = (const float*)d_in[11];
    const float* lrw  = (const float*)d_in[12];
    const float* lrb  = (const float*)d_in[13];
    const float* chw  = (const float*)d_in[14];
    const float* chb  = (const float*)d_in[15];
    const float* lhw  = (const float*)d_in[16];
    const float* lhb  = (const float*)d_in[17];
    const float* ow   = (const float*)d_in[18];
    const float* ob   = (const float*)d_in[19];
    float* out = (float*)d_out;

    const int n = in_sizes[0] / FT;     // 100000
    const int e = in_sizes[2];          // 1600000
    const int* src = ei;
    const int* dst = ei + e;

    float* wx  = (float*)d_ws;                  // n*96
    float* agg = wx + (size_t)n * FT;           // n*96
    float* dis = agg + (size_t)n * FT;          // n   (deg -> rsqrt(deg) in place)

    k_deg_init <<<(n + 255) / 256, 256, 0, stream>>>(dis, n);
    k_deg_accum<<<(e + 255) / 256, 256, 0, stream>>>(dst, ew, dis, e);
    k_dis      <<<(n + 255) / 256, 256, 0, stream>>>(dis, n);
    k_fwx      <<<(n + 255) / 256, 256, 0, stream>>>(x, mlpw, mlpb, wx, n);
    {
        long long tot = (long long)n * FT;
        k_selfinit<<<(int)((tot + 255) / 256), 256, 0, stream>>>(wx, dis, agg, n);
    }
    k_edge<<<(e + 7) / 8, 256, 0, stream>>>(src, dst, ew, dis, wx, agg, e);
    k_gru <<<(n + NPB - 1) / NPB, GRU_THREADS, 0, stream>>>(
        agg, att, czw, czb, lzw, lzb, crw, crb, lrw, lrb, chw, chb, lhw, lhb,
        ow, ob, out, n);
}